// ChamferDistLoss_42820823941122
// MI455X (gfx1250) — compile-verified
//
#include <hip/hip_runtime.h>

typedef float v2f __attribute__((ext_vector_type(2)));
typedef float v8f __attribute__((ext_vector_type(8)));

#define ROWS_PER_WAVE   32    // two 16-row A fragments per wave
#define WAVES_PER_BLOCK 8
#define ROWS_PER_BLOCK  256   // 8 waves * 32 rows
#define CHUNK           1024  // target points staged in LDS per pass (12 KB)
#define TILE            16

// For each query point q (per batch) compute min over targets t of
// (|t|^2 - 2 q.t)  -- the |q|^2 term is constant under the min and is added
// globally by the sum-of-squares kernel. One partial sum per block.
__global__ __launch_bounds__(256) void chamfer_dir_kernel(
    const float* __restrict__ Q, const float* __restrict__ T,
    int nQ, int nT, float* __restrict__ partials)
{
    __shared__ float sT[CHUNK * 3];
    __shared__ float waveSums[WAVES_PER_BLOCK];

    const int blocksPerBatch = nQ / ROWS_PER_BLOCK;
    const int b    = blockIdx.x / blocksPerBatch;
    const int rb   = blockIdx.x % blocksPerBatch;
    const int tid  = threadIdx.x;
    const int wave = tid >> 5;
    const int lane = tid & 31;
    const int half = lane >> 4;   // WMMA half-wave: K-split for A/B fragments
    const int l15  = lane & 15;

    // ---- Two A fragments: 32 query rows, A[m] = (-2x, -2y, -2z, 1) ----
    // 32-bit A 16x4 layout: lanes 0-15 hold K={0,1}, lanes 16-31 hold K={2,3}
    const int row0 = rb * ROWS_PER_BLOCK + wave * ROWS_PER_WAVE + l15;
    const float* qp0 = Q + ((size_t)b * nQ + row0) * 3;
    const float* qp1 = qp0 + 16 * 3;
    const float q0x = qp0[0], q0y = qp0[1], q0z = qp0[2];
    const float q1x = qp1[0], q1y = qp1[1], q1z = qp1[2];
    v2f afrag0, afrag1;
    afrag0.x = half ? (-2.0f * q0z) : (-2.0f * q0x);
    afrag0.y = half ? 1.0f          : (-2.0f * q0y);
    afrag1.x = half ? (-2.0f * q1z) : (-2.0f * q1x);
    afrag1.y = half ? 1.0f          : (-2.0f * q1y);

    v8f acc0, acc1;
    #pragma unroll
    for (int r = 0; r < 8; ++r) { acc0[r] = 3.0e38f; acc1[r] = 3.0e38f; }
    const v8f czero = {0.f, 0.f, 0.f, 0.f, 0.f, 0.f, 0.f, 0.f};

    const float* tbase = T + (size_t)b * nT * 3;

    for (int c0 = 0; c0 < nT; c0 += CHUNK) {
        // Stage CHUNK target points (xyz) into LDS: 3072 floats = 768 float4
        {
            const float4* src = (const float4*)(tbase + (size_t)c0 * 3);
            float4* dst = (float4*)sT;
            #pragma unroll
            for (int k = 0; k < 3; ++k)
                dst[tid + k * 256] = src[tid + k * 256];
        }
        __syncthreads();

        #pragma unroll 2
        for (int jt = 0; jt < CHUNK / TILE; ++jt) {
            const int n = jt * TILE + l15;
            const float tx = sT[n * 3 + 0];
            const float ty = sT[n * 3 + 1];
            const float tz = sT[n * 3 + 2];
            const float nrm = __builtin_fmaf(tx, tx, __builtin_fmaf(ty, ty, tz * tz));
            // B fragment: B[:,n] = (x', y', z', |t|^2); same K-split as A
            v2f bfrag;
            bfrag.x = half ? tz  : tx;
            bfrag.y = half ? nrm : ty;
            // Two 16x16 tiles per B fragment: amortize the B build + LDS reads
            v8f d0 = __builtin_amdgcn_wmma_f32_16x16x4_f32(
                false, afrag0, false, bfrag, (short)0, czero, false, false);
            v8f d1 = __builtin_amdgcn_wmma_f32_16x16x4_f32(
                false, afrag1, false, bfrag, (short)0, czero, false, false);
            #pragma unroll
            for (int r = 0; r < 8; ++r) {
                acc0[r] = fminf(acc0[r], d0[r]);
                acc1[r] = fminf(acc1[r], d1[r]);
            }
        }
        __syncthreads();
    }

    // ---- min across the 16 columns held in each half-wave ----
    #pragma unroll
    for (int mask = 1; mask <= 8; mask <<= 1) {
        #pragma unroll
        for (int r = 0; r < 8; ++r) {
            acc0[r] = fminf(acc0[r], __shfl_xor(acc0[r], mask, 32));
            acc1[r] = fminf(acc1[r], __shfl_xor(acc1[r], mask, 32));
        }
    }
    // half 0 lanes now hold row-mins for rows {0-7,16-23}, half 1 for {8-15,24-31}
    float tsum = 0.f;
    #pragma unroll
    for (int r = 0; r < 8; ++r) tsum += acc0[r] + acc1[r];
    tsum += __shfl_xor(tsum, 16, 32);  // combine the two halves -> 32-row sum

    if (lane == 0) waveSums[wave] = tsum;
    __syncthreads();
    if (tid == 0) {
        float s = 0.f;
        #pragma unroll
        for (int w = 0; w < WAVES_PER_BLOCK; ++w) s += waveSums[w];
        partials[blockIdx.x] = s;
    }
}

// Sum of squares of all coordinates of both clouds (the dropped |q|^2 terms).
__global__ __launch_bounds__(256) void sumsq_kernel(
    const float* __restrict__ a, int na,
    const float* __restrict__ b, int nb,
    float* __restrict__ partials)
{
    __shared__ float red[256];
    const int stride = 256 * gridDim.x;
    float s = 0.f;
    for (int i = blockIdx.x * 256 + threadIdx.x; i < na; i += stride) {
        float v = a[i]; s = __builtin_fmaf(v, v, s);
    }
    for (int i = blockIdx.x * 256 + threadIdx.x; i < nb; i += stride) {
        float v = b[i]; s = __builtin_fmaf(v, v, s);
    }
    red[threadIdx.x] = s;
    __syncthreads();
    for (int off = 128; off > 0; off >>= 1) {
        if (threadIdx.x < off) red[threadIdx.x] += red[threadIdx.x + off];
        __syncthreads();
    }
    if (threadIdx.x == 0) partials[blockIdx.x] = red[0];
}

// Single-block deterministic final sum of all partials -> scalar loss.
__global__ __launch_bounds__(256) void final_sum_kernel(
    const float* __restrict__ p, int n, float* __restrict__ out)
{
    __shared__ float red[256];
    float s = 0.f;
    for (int i = threadIdx.x; i < n; i += 256) s += p[i];
    red[threadIdx.x] = s;
    __syncthreads();
    for (int off = 128; off > 0; off >>= 1) {
        if (threadIdx.x < off) red[threadIdx.x] += red[threadIdx.x + off];
        __syncthreads();
    }
    if (threadIdx.x == 0) out[0] = red[0];
}

extern "C" void kernel_launch(void* const* d_in, const int* in_sizes, int n_in,
                              void* d_out, int out_size, void* d_ws, size_t ws_size,
                              hipStream_t stream)
{
    (void)n_in; (void)out_size; (void)ws_size;
    const float* inp  = (const float*)d_in[0];   // [B, N, 3]
    const float* outp = (const float*)d_in[1];   // [B, M, 3]
    const int B = 8;
    const int N = in_sizes[0] / (B * 3);
    const int M = in_sizes[1] / (B * 3);

    float* ws = (float*)d_ws;
    const int blocks0 = B * (N / ROWS_PER_BLOCK);  // 256
    const int blocks1 = B * (M / ROWS_PER_BLOCK);  // 256
    const int nbNorm  = 128;

    // Direction 1: for each input point, min distance into output cloud
    chamfer_dir_kernel<<<blocks0, 256, 0, stream>>>(inp, outp, N, M, ws);
    // Direction 2: swapped roles (chamfer symmetry) -> no cross-block atomics
    chamfer_dir_kernel<<<blocks1, 256, 0, stream>>>(outp, inp, M, N, ws + blocks0);
    // The |q|^2 terms dropped from the min loops
    sumsq_kernel<<<nbNorm, 256, 0, stream>>>(inp, in_sizes[0], outp, in_sizes[1],
                                             ws + blocks0 + blocks1);
    final_sum_kernel<<<1, 256, 0, stream>>>(ws, blocks0 + blocks1 + nbNorm,
                                            (float*)d_out);
}